// EncoderDecoder_25288767439278
// MI455X (gfx1250) — compile-verified
//
#include <hip/hip_runtime.h>
#include <hip/hip_bf16.h>
#include <math.h>

// Problem constants (from reference)
#define NN      50000
#define LATENT  16
#define KNB     16
#define CC      64
#define BB      4
#define H1      200
#define NTILES  13      // ceil(200/16)

typedef float v2f __attribute__((ext_vector_type(2)));
typedef float v8f __attribute__((ext_vector_type(8)));

// ---------------------------------------------------------------------------
// Kernel A: split-K GEMM1 via V_WMMA_F32_16X16X4_F32.
//   P[slice][tile][lane][v] partial of (x_pad(16xK) @ enc1_w.T(Kx208)) tile.
//   Padded A rows (m>=4) / B cols (o>=200) read CLAMPED rows: their D cells
//   are garbage but never consumed by the reducer, so no masking is needed.
// One wave per block; grid = (13 tiles, used slices). Branch-free hot loop.
// ---------------------------------------------------------------------------
__global__ void gemm1_wmma_kernel(const float* __restrict__ x,
                                  const float* __restrict__ w1,
                                  float* __restrict__ P,
                                  int kslice) {
  const int tile = blockIdx.x;       // 0..12
  const int slice = blockIdx.y;
  const int lane = threadIdx.x;      // 0..31 (wave32)
  const int nIdx = lane & 15;
  const int hf   = lane >> 4;        // 0 or 1
  const int o    = tile * 16 + nIdx; // output column (0..207)
  const int m    = nIdx;             // A-matrix row (batch, padded)
  const int mc   = m < BB ? m : (BB - 1);     // clamp: padded rows re-read row 3
  const int oc   = o < H1 ? o : (H1 - 1);     // clamp: padded cols re-read col 199

  const float* xrow = x  + (size_t)mc * NN + 2 * hf;
  const float* wrow = w1 + (size_t)oc * NN + 2 * hf;

  const int j0 = slice * kslice;
  int jend = j0 + kslice;
  if (jend > NN) jend = NN;          // NN % 4 == 0, j0 % 32 == 0

  v8f c = {};
  int j = j0;
  // main loop: 32 K-values (8 WMMAs) per group, prefetch weight stream ahead
  for (; j + 32 <= jend; j += 32) {
    __builtin_prefetch(wrow + j + 512, 0, 1);   // 2KB ahead on the 40MB stream
#pragma unroll
    for (int u = 0; u < 8; ++u) {
      const v2f a = *(const v2f*)(xrow + j + 4 * u);
      const v2f b = *(const v2f*)(wrow + j + 4 * u);
      c = __builtin_amdgcn_wmma_f32_16x16x4_f32(
          false, a, false, b, (short)0, c, false, false);
    }
  }
  // remainder (last slice only; multiples of 4)
  for (; j < jend; j += 4) {
    const v2f a = *(const v2f*)(xrow + j);
    const v2f b = *(const v2f*)(wrow + j);
    c = __builtin_amdgcn_wmma_f32_16x16x4_f32(
        false, a, false, b, (short)0, c, false, false);
  }

  float* dst = P + (((size_t)slice * NTILES + tile) * 32 + lane) * 8;
  *(v8f*)dst = c;   // element (M=v+8*hf, N=nIdx) at [..][lane][v]
}

// ---------------------------------------------------------------------------
// Kernel B: reduce split-K partials, add bias, sigmoid -> h (4 x 200).
// For M=b<4: lane = n, v = b.
// ---------------------------------------------------------------------------
__global__ void reduce_h_kernel(const float* __restrict__ P,
                                const float* __restrict__ b1,
                                float* __restrict__ h,
                                int slices) {
  const int idx = blockIdx.x * blockDim.x + threadIdx.x;
  if (idx >= BB * H1) return;
  const int b = idx / H1;
  const int o = idx % H1;
  const int tile = o >> 4;
  const int n = o & 15;
  float s = 0.0f;
  for (int sl = 0; sl < slices; ++sl)
    s += P[(((size_t)sl * NTILES + tile) * 32 + n) * 8 + b];
  s += b1[o];
  h[b * H1 + o] = 1.0f / (1.0f + __expf(-s));
}

// ---------------------------------------------------------------------------
// Kernel C: tiny MLP chain (single 256-thread block, ~0.3 MFLOP total).
//   encoded(4x16) -> silu MLP 200 -> 200 -> 64 -> Hc = sigmoid(0.01*z)
// ---------------------------------------------------------------------------
__global__ void mlp_small_kernel(const float* __restrict__ h,
                                 const float* __restrict__ e2w, const float* __restrict__ e2b,
                                 const float* __restrict__ h0w, const float* __restrict__ h0b,
                                 const float* __restrict__ h1w, const float* __restrict__ h1b,
                                 const float* __restrict__ h2w, const float* __restrict__ h2b,
                                 float* __restrict__ enc,
                                 float* __restrict__ z0,
                                 float* __restrict__ z1,
                                 float* __restrict__ Hc) {
  const int t = threadIdx.x;
  // encoded = h @ enc2_w.T + enc2_b   (4 x 16)
  if (t < BB * LATENT) {
    const int b = t >> 4, n = t & 15;
    float s = e2b[n];
    for (int j = 0; j < H1; ++j) s += h[b * H1 + j] * e2w[n * H1 + j];
    enc[t] = s;
  }
  __syncthreads();
  // z0 = silu(encoded @ h0_w.T + h0_b)   (4 x 200)
  for (int idx = t; idx < BB * H1; idx += 256) {
    const int b = idx / H1, o = idx % H1;
    float s = h0b[o];
    for (int j = 0; j < LATENT; ++j) s += enc[b * LATENT + j] * h0w[o * LATENT + j];
    const float sg = 1.0f / (1.0f + __expf(-s));
    z0[idx] = s * sg;
  }
  __syncthreads();
  // z1 = silu(z0 @ h1_w.T + h1_b)   (4 x 200)
  for (int idx = t; idx < BB * H1; idx += 256) {
    const int b = idx / H1, o = idx % H1;
    float s = h1b[o];
    for (int j = 0; j < H1; ++j) s += z0[b * H1 + j] * h1w[o * H1 + j];
    const float sg = 1.0f / (1.0f + __expf(-s));
    z1[idx] = s * sg;
  }
  __syncthreads();
  // Hc = sigmoid(0.01 * (z1 @ h2_w.T + h2_b))   (4 x 64)
  for (int idx = t; idx < BB * CC; idx += 256) {
    const int b = idx >> 6, c = idx & 63;
    float s = h2b[c];
    for (int j = 0; j < H1; ++j) s += z1[b * H1 + j] * h2w[c * H1 + j];
    Hc[idx] = 1.0f / (1.0f + __expf(-0.01f * s));
  }
}

// ---------------------------------------------------------------------------
// Kernel D: per-node windowed neighbour convolution + final contraction.
// One thread per node. dec_w (3.2 MB) is L2-resident -> gathers hit L2.
// ---------------------------------------------------------------------------
__global__ void node_kernel(const int*   __restrict__ nbr_id,
                            const float* __restrict__ nbr_d,
                            const int*   __restrict__ labels,
                            const float* __restrict__ dec_w,
                            const float* __restrict__ dec_b,
                            const float* __restrict__ enc,
                            const float* __restrict__ Hc,
                            const float* __restrict__ Bsc,
                            float* __restrict__ out) {
  const int i = blockIdx.x * blockDim.x + threadIdx.x;
  if (i >= NN) return;

  const int lab = labels[i];
  const float B0 = Bsc[0];

  float invd[BB][LATENT], conv[BB][LATENT], ssum[BB][LATENT];
#pragma unroll
  for (int b = 0; b < BB; ++b) {
    const float Hv = Hc[b * CC + lab];
    const float r = 1.0f - 0.5f * Hv;   // (1 - H/2)
    float w = B0;                       // (1-H/2)^0 * B
#pragma unroll
    for (int n = 0; n < LATENT; ++n) {
      const float d = 2.0f * w;         // MU = 2
      invd[b][n] = 1.0f / (d * d);
      conv[b][n] = 0.0f;
      ssum[b][n] = 0.0f;
      w *= r;
    }
  }

  const int*   ids = nbr_id + (size_t)i * KNB;
  const float* dst = nbr_d  + (size_t)i * KNB;

  for (int k = 0; k < KNB; ++k) {
    const int id = ids[k];
    const float dd = dst[k];
    const float d2 = dd * dd;
    const float4* row = (const float4*)(dec_w + (size_t)id * LATENT);
    const float4 r0 = row[0], r1 = row[1], r2 = row[2], r3 = row[3];
    const float basis[LATENT] = {r0.x, r0.y, r0.z, r0.w,
                                 r1.x, r1.y, r1.z, r1.w,
                                 r2.x, r2.y, r2.z, r2.w,
                                 r3.x, r3.y, r3.z, r3.w};
#pragma unroll
    for (int b = 0; b < BB; ++b) {
#pragma unroll
      for (int n = 0; n < LATENT; ++n) {
        float win = 1.0f - d2 * invd[b][n];
        win = win > 0.0f ? win : 0.0f;
        ssum[b][n] += win;
        conv[b][n] += win * basis[n];
      }
    }
  }

  const float db = dec_b[i];
#pragma unroll
  for (int b = 0; b < BB; ++b) {
    float acc = db;
#pragma unroll
    for (int n = 0; n < LATENT; ++n)
      acc += enc[b * LATENT + n] * (conv[b][n] / ssum[b][n]);
    out[(size_t)b * NN + i] = acc;
  }
}

// ---------------------------------------------------------------------------
extern "C" void kernel_launch(void* const* d_in, const int* in_sizes, int n_in,
                              void* d_out, int out_size, void* d_ws, size_t ws_size,
                              hipStream_t stream) {
  const float* x      = (const float*)d_in[0];
  const int*   nbr_id = (const int*)  d_in[1];
  const float* nbr_d  = (const float*)d_in[2];
  const int*   labels = (const int*)  d_in[3];
  const float* enc1_w = (const float*)d_in[4];
  const float* enc1_b = (const float*)d_in[5];
  const float* enc2_w = (const float*)d_in[6];
  const float* enc2_b = (const float*)d_in[7];
  const float* dec_w  = (const float*)d_in[8];
  const float* dec_b  = (const float*)d_in[9];
  const float* h0_w   = (const float*)d_in[10];
  const float* h0_b   = (const float*)d_in[11];
  const float* h1_w   = (const float*)d_in[12];
  const float* h1_b   = (const float*)d_in[13];
  const float* h2_w   = (const float*)d_in[14];
  const float* h2_b   = (const float*)d_in[15];
  const float* Bsc    = (const float*)d_in[16];
  float* out = (float*)d_out;
  float* ws  = (float*)d_ws;

  // Workspace budget -> split-K slice count (deterministic, adapts to ws_size)
  const size_t reserveF = (size_t)(BB*H1 + BB*LATENT + BB*H1 + BB*H1 + BB*CC); // 2720
  long availF = (long)(ws_size / sizeof(float)) - (long)reserveF;
  int slices = (int)(availF / (NTILES * 256));
  if (slices < 1)  slices = 1;
  if (slices > 64) slices = 64;
  int kslice = ((NN + slices - 1) / slices + 31) & ~31;     // multiple of 32
  int used   = (NN + kslice - 1) / kslice;                  // actual slices used

  float* P   = ws;
  float* h   = ws + (size_t)slices * NTILES * 256;
  float* enc = h + BB * H1;
  float* z0  = enc + BB * LATENT;
  float* z1  = z0 + BB * H1;
  float* Hc  = z1 + BB * H1;

  gemm1_wmma_kernel<<<dim3(NTILES, used), 32, 0, stream>>>(x, enc1_w, P, kslice);
  reduce_h_kernel<<<dim3((BB * H1 + 255) / 256), 256, 0, stream>>>(P, enc1_b, h, used);
  mlp_small_kernel<<<1, 256, 0, stream>>>(h, enc2_w, enc2_b,
                                          h0_w, h0_b, h1_w, h1_b, h2_w, h2_b,
                                          enc, z0, z1, Hc);
  node_kernel<<<dim3((NN + 255) / 256), 256, 0, stream>>>(nbr_id, nbr_d, labels,
                                                          dec_w, dec_b, enc, Hc, Bsc, out);
}